// MemoryAugmentation_38568806318371
// MI455X (gfx1250) — compile-verified
//
#include <hip/hip_runtime.h>
#include <hip/hip_bf16.h>

// Problem constants (from reference): x[32,256,88,88], mem[10,88,88]
#define RDIM   8192      // B*C rows
#define KDIM   7744      // H*W
#define MSLOT  10
#define TILE_R 16        // rows per block == WMMA M/N
#define CHUNK  704       // K chunk (11 * 704 = 7744 exactly)
#define LDS_STRIDE 708   // CHUNK + 4 dwords pad -> conflict-free A-frag reads
#define NCHUNK 11
#define BLOCK  256
#define NWAVE  8
#define STEPS_PER_WAVE 22   // (CHUNK/4) / NWAVE

typedef __attribute__((ext_vector_type(2))) float v2f;
typedef __attribute__((ext_vector_type(8))) float v8f;

__global__ __launch_bounds__(BLOCK)
void memaug_fused_kernel(const float* __restrict__ x,
                         const float* __restrict__ mem,
                         float* __restrict__ out)
{
    // double-buffered x tile stage (async DMA target), padded stride
    __shared__ __align__(16) float lds_x[2][TILE_R * LDS_STRIDE];   // 2 x 45.3 KB
    __shared__ float redbuf[NWAVE * 32 * 8];                        // 8 KB
    __shared__ float score_raw[TILE_R * 16];
    __shared__ float score_s[TILE_R * MSLOT];

    const int tid  = threadIdx.x;
    const int lane = tid & 31;
    const int wv   = tid >> 5;
    const int r0   = blockIdx.x * TILE_R;

    // WMMA fragment lane mapping (32-bit 16x4 A / 4x16 B layouts):
    //   lanes 0-15: M/N = lane,   K sub-offset 0; lanes 16-31: M/N = lane-16, K+2
    const int frag_mn = lane & 15;
    const int khalf   = (lane >> 4) * 2;
    // score columns 10..15 are never read -> clamp row, no zeroing needed
    const int brow    = (frag_mn < MSLOT) ? frag_mn : 0;

    // cooperative stage mapping: 16 threads per row, 11 x b128 per thread
    const int ld_row = tid >> 4;
    const int ld_c0  = tid & 15;

    const float* gsrc_row = x + (size_t)(r0 + ld_row) * KDIM + ld_c0 * 4;
    // LDS byte address = low 32 bits of generic shared pointer (ISA aperture rule)
    const unsigned lds_d0 =
        (unsigned)(size_t)(&lds_x[0][ld_row * LDS_STRIDE]) + ld_c0 * 16;
    const unsigned lds_d1 =
        (unsigned)(size_t)(&lds_x[1][ld_row * LDS_STRIDE]) + ld_c0 * 16;

    // async global -> LDS DMA of one 16 x 704 chunk (per-lane b128 copies)
    auto stage = [&](int ch) {
        const float* src = gsrc_row + ch * CHUNK;
        unsigned     dst = (ch & 1) ? lds_d1 : lds_d0;
        #pragma unroll
        for (int j = 0; j < 11; ++j) {
            asm volatile("global_load_async_to_lds_b128 %0, %1, off"
                         :: "v"(dst), "v"(src) : "memory");
            src += 64;    // 16 lanes * 4 floats
            dst += 256;   // bytes
        }
    };

    v8f acc = {0.f, 0.f, 0.f, 0.f, 0.f, 0.f, 0.f, 0.f};

    stage(0);   // prologue fill of buffer 0

    for (int ch = 0; ch < NCHUNK; ++ch) {
        asm volatile("s_wait_asynccnt 0x0" ::: "memory");  // my DMA landed
        __syncthreads();                                   // everyone's landed / reads done
        if (ch + 1 < NCHUNK) stage(ch + 1);                // overlap next DMA with WMMA
        if (ch + 2 < NCHUNK)
            __builtin_prefetch(gsrc_row + (size_t)(ch + 2) * CHUNK, 0, 3);

        const float* abuf = &lds_x[ch & 1][frag_mn * LDS_STRIDE + khalf];
        const float* bptr = mem + (size_t)brow * KDIM + ch * CHUNK + khalf;
        const int kbase = wv * (STEPS_PER_WAVE * 4);
        #pragma unroll
        for (int s = 0; s < STEPS_PER_WAVE; ++s) {
            const int kk = kbase + s * 4;
            v2f a = *(const v2f*)(abuf + kk);   // 16 x-rows x 4K, conflict-free
            v2f b = *(const v2f*)(bptr + kk);   // mem^T 4K x 16 (L2-resident)
            acc = __builtin_amdgcn_wmma_f32_16x16x4_f32(
                      false, a, false, b, (short)0, acc, false, false);
        }
    }

    // ---- cross-wave reduction of the 16x16 partial accumulators ----
    #pragma unroll
    for (int i = 0; i < 8; ++i)
        redbuf[i * 256 + wv * 32 + lane] = acc[i];
    __syncthreads();

    if (wv == 0) {
        #pragma unroll
        for (int i = 0; i < 8; ++i) {
            float ssum = 0.f;
            #pragma unroll
            for (int w2 = 0; w2 < NWAVE; ++w2)
                ssum += redbuf[i * 256 + w2 * 32 + lane];
            // D layout: vgpr i holds M=i (lanes 0-15) / M=i+8 (lanes 16-31), N=lane&15
            const int Mrow = i + 8 * (lane >> 4);
            const int Ncol = lane & 15;
            score_raw[Mrow * 16 + Ncol] = ssum;
        }
    }
    __syncthreads();

    // ---- softmax over the 10 slots, one row per thread ----
    if (tid < TILE_R) {
        float v[MSLOT];
        float mx = -3.402823466e38f;
        #pragma unroll
        for (int m = 0; m < MSLOT; ++m) {
            v[m] = score_raw[tid * 16 + m];
            mx = fmaxf(mx, v[m]);
        }
        float ssum = 0.f;
        #pragma unroll
        for (int m = 0; m < MSLOT; ++m) { v[m] = __expf(v[m] - mx); ssum += v[m]; }
        const float inv = 1.f / ssum;
        #pragma unroll
        for (int m = 0; m < MSLOT; ++m) score_s[tid * MSLOT + m] = v[m] * inv;
    }
    __syncthreads();

    // ---- phase 2: value[row][col] = sum_m score[row][m] * mem[m][col] ----
    // each wave produces 2 rows; scores in registers; fully coalesced IO
    const int ra = 2 * wv, rb = 2 * wv + 1;
    float sa[MSLOT], sb[MSLOT];
    #pragma unroll
    for (int m = 0; m < MSLOT; ++m) {
        sa[m] = score_s[ra * MSLOT + m];
        sb[m] = score_s[rb * MSLOT + m];
    }
    float* outa = out + (size_t)(r0 + ra) * KDIM;
    float* outb = out + (size_t)(r0 + rb) * KDIM;
    #pragma unroll 1
    for (int j = 0; j < KDIM / 64; ++j) {        // 121 iters, float2 per lane
        const int col = (j * 32 + lane) * 2;
        float va0 = 0.f, va1 = 0.f, vb0 = 0.f, vb1 = 0.f;
        #pragma unroll
        for (int m = 0; m < MSLOT; ++m) {
            v2f mm = *(const v2f*)(mem + (size_t)m * KDIM + col);
            va0 += sa[m] * mm.x;  va1 += sa[m] * mm.y;
            vb0 += sb[m] * mm.x;  vb1 += sb[m] * mm.y;
        }
        v2f va = {va0, va1}, vb = {vb0, vb1};
        *(v2f*)(outa + col) = va;
        *(v2f*)(outb + col) = vb;
    }
}

extern "C" void kernel_launch(void* const* d_in, const int* in_sizes, int n_in,
                              void* d_out, int out_size, void* d_ws, size_t ws_size,
                              hipStream_t stream) {
    (void)in_sizes; (void)n_in; (void)d_ws; (void)ws_size; (void)out_size;
    const float* x   = (const float*)d_in[0];   // [32,256,88,88] fp32
    const float* mem = (const float*)d_in[1];   // [10,88,88] fp32
    float* out = (float*)d_out;                 // [32,256,88,88] fp32
    dim3 grid(RDIM / TILE_R);                   // 512 blocks
    dim3 block(BLOCK);                          // 8 waves
    memaug_fused_kernel<<<grid, block, 0, stream>>>(x, mem, out);
}